// GraphNetHighCapacity_83614423318870
// MI455X (gfx1250) — compile-verified
//
#include <hip/hip_runtime.h>
#include <hip/hip_bf16.h>
#include <float.h>
#include <math.h>

// ---------------------------------------------------------------------------
// GCN forward for MI455X (gfx1250, wave32). fp32 everywhere (network is
// memory/L2-bound; the 192MB L2 holds the 80MB layer-4 feature matrix, so
// lower precision buys nothing). Dense GEMMs use V_WMMA_F32_16X16X4_F32 with
// a guard-free inner loop (M%16==0, K%4==0, Nc%16==0 enforced by padding both
// x and W1 from K=6 to K=8).
// ---------------------------------------------------------------------------

typedef __attribute__((ext_vector_type(2))) float v2f;
typedef __attribute__((ext_vector_type(8))) float v8f;

#define BN_EPS 1e-5f

// ---------------- WMMA fp32 GEMM: C[M,Nc] = A[M,K] @ W[K,Nc] ----------------
// Requires: M % 16 == 0, K % 4 == 0, Nc % 16 == 0 (caller guarantees).
// One wave per 16x16 C tile; 4 waves per block along N.
// A fragment (16x4 f32, ISA 7.12.2): lanes 0-15 hold K=0 (VGPR0), K=1 (VGPR1);
// lanes 16-31 hold K=2,3. B (4x16) mirrored across lanes. C/D: 8 VGPRs,
// lanes 0-15 -> M=r, lanes 16-31 -> M=r+8.
__global__ void gemm_wmma_f32(const float* __restrict__ A,
                              const float* __restrict__ W,
                              float* __restrict__ C,
                              int M, int K, int Nc)
{
    const int lane   = threadIdx.x;       // 0..31
    const int l      = lane & 15;
    const int half   = lane >> 4;         // 0 or 1
    const int ntile  = Nc >> 4;
    const int tile_n = blockIdx.y * blockDim.y + threadIdx.y;
    if (tile_n >= ntile) return;          // wave-uniform exit: EXEC stays full
    const int row0 = blockIdx.x << 4;
    const int col0 = tile_n << 4;
    const int m = row0 + l;
    const int n = col0 + l;

    // Per-lane streaming pointers; (half<<1) is even and K is even, so the
    // v2f (b64) load of A is always 8-byte aligned.
    const float* __restrict__ Ap = A + (size_t)m * K + (half << 1);
    const float* __restrict__ Wp = W + (size_t)(half << 1) * Nc + n;

    v8f acc = {};
#pragma unroll 4
    for (int kb = 0; kb < K; kb += 4) {
        v2f a = *(const v2f*)Ap;          // A[m][kb+2*half], A[m][kb+2*half+1]
        v2f b;
        b.x = Wp[0];                      // W[kb+2*half  ][n]
        b.y = Wp[Nc];                     // W[kb+2*half+1][n]
        Ap += 4;
        Wp += (size_t)4 * Nc;
        // 8 args: (neg_a, A, neg_b, B, c_mod, C, reuse_a, reuse_b)
        acc = __builtin_amdgcn_wmma_f32_16x16x4_f32(
            false, a, false, b, (short)0, acc, false, false);
    }

    float* __restrict__ Cp = C + (size_t)(row0 + (half << 3)) * Nc + n;
#pragma unroll
    for (int r = 0; r < 8; ++r)
        Cp[(size_t)r * Nc] = acc[r];
}

// ---------------- padding (layer 1: K=6 -> K=8, no K guard in GEMM) ---------
__global__ void pad_x(const float* __restrict__ x, float* __restrict__ xp, int n)
{
    int idx = blockIdx.x * blockDim.x + threadIdx.x;
    if (idx >= n * 8) return;
    int i = idx >> 3, f = idx & 7;
    xp[idx] = (f < 6) ? x[i * 6 + f] : 0.0f;
}

__global__ void pad_w1(const float* __restrict__ w1, float* __restrict__ w1p)
{
    int idx = blockIdx.x * blockDim.x + threadIdx.x;   // [8,32]
    if (idx >= 8 * 32) return;
    int k = idx >> 5;
    w1p[idx] = (k < 6) ? w1[idx] : 0.0f;
}

// ---------------- degree / normalization ----------------
__global__ void deg_init(float* __restrict__ deg, int n)
{
    int i = blockIdx.x * blockDim.x + threadIdx.x;
    if (i < n) deg[i] = 1.0f;             // self-loop
}

__global__ void deg_count(const int* __restrict__ dst, float* __restrict__ deg, int e)
{
    int i = blockIdx.x * blockDim.x + threadIdx.x;
    if (i < e) atomicAdd(&deg[dst[i]], 1.0f);
}

__global__ void deg_rsqrt(const float* __restrict__ deg, float* __restrict__ dinv, int n)
{
    int i = blockIdx.x * blockDim.x + threadIdx.x;
    if (i < n) dinv[i] = rsqrtf(deg[i]);
}

// agg[i] = lin[i] / deg[i]  (self-loop term: dinv[i]^2 == 1/deg[i]); float4.
__global__ void selfloop_init4(const float4* __restrict__ lin,
                               const float* __restrict__ deg,
                               float4* __restrict__ agg, int n, int F4)
{
    long long idx = (long long)blockIdx.x * blockDim.x + threadIdx.x;
    if (idx >= (long long)n * F4) return;
    int i = (int)(idx / F4);
    float s = 1.0f / deg[i];
    float4 v = lin[idx];
    agg[idx] = make_float4(v.x * s, v.y * s, v.z * s, v.w * s);
}

// agg[dst] += dinv[src]*dinv[dst]*lin[src] over (edge, 4-feature-chunk) pairs.
__global__ void edge_scatter4(const int* __restrict__ src,
                              const int* __restrict__ dst,
                              const float* __restrict__ dinv,
                              const float4* __restrict__ lin,
                              float* __restrict__ agg, int e, int F4)
{
    long long idx = (long long)blockIdx.x * blockDim.x + threadIdx.x;
    if (idx >= (long long)e * F4) return;
    int ei = (int)(idx / F4);
    int c  = (int)(idx % F4);
    int s = src[ei], d = dst[ei];
    float w = dinv[s] * dinv[d];
    float4 v = lin[(size_t)s * F4 + c];
    float* ap = agg + ((size_t)d * F4 + c) * 4;
    atomicAdd(ap + 0, w * v.x);
    atomicAdd(ap + 1, w * v.y);
    atomicAdd(ap + 2, w * v.z);
    atomicAdd(ap + 3, w * v.w);
}

// ---------------- BatchNorm (training mode, biased variance) ----------------
__global__ void bn_stats(const float* __restrict__ h,
                         float* __restrict__ mean, float* __restrict__ var,
                         int rows, int F)
{
    const int feat = blockIdx.x;          // one block per feature
    const int tid  = threadIdx.x;
    float s = 0.0f, ss = 0.0f;
    for (int i = tid; i < rows; i += blockDim.x) {
        float v = h[(size_t)i * F + feat];
        s += v; ss += v * v;
    }
    __shared__ float sh[256], shh[256];
    sh[tid] = s; shh[tid] = ss;
    __syncthreads();
    for (int st = blockDim.x >> 1; st > 0; st >>= 1) {
        if (tid < st) { sh[tid] += sh[tid + st]; shh[tid] += shh[tid + st]; }
        __syncthreads();
    }
    if (tid == 0) {
        float m = sh[0] / rows;
        mean[feat] = m;
        var[feat]  = shh[0] / rows - m * m;
    }
}

// In-place gamma*(x-m)*rsqrt(v+eps)+beta, then ReLU. The reference's pre-BN
// bias b is a per-feature constant exactly cancelled by mean subtraction, so
// it is (correctly) omitted.
__global__ void bn_relu(float* __restrict__ h,
                        const float* __restrict__ mean,
                        const float* __restrict__ var,
                        const float* __restrict__ g,
                        const float* __restrict__ be,
                        int rows, int F)
{
    long long idx = (long long)blockIdx.x * blockDim.x + threadIdx.x;
    if (idx >= (long long)rows * F) return;
    int f = (int)(idx % F);
    float y = g[f] * (h[idx] - mean[f]) * rsqrtf(var[f] + BN_EPS) + be[f];
    h[idx] = fmaxf(y, 0.0f);
}

// ---------------- global max pool ----------------
__global__ void pool_zero(float* __restrict__ p, int n)
{
    int i = blockIdx.x * blockDim.x + threadIdx.x;
    if (i < n) p[i] = 0.0f;
}

// h is post-ReLU (>= 0), so uint compare == float compare; every graph is
// non-empty w.h.p. so max >= 0 matches segment_max.
__global__ void pool_max4(const float4* __restrict__ h,
                          const int* __restrict__ batch,
                          float* __restrict__ p, int n, int F4)
{
    long long idx = (long long)blockIdx.x * blockDim.x + threadIdx.x;
    if (idx >= (long long)n * F4) return;
    int i = (int)(idx / F4);
    int c = (int)(idx % F4);
    float4 v = h[idx];
    unsigned int* pp = (unsigned int*)(p + ((size_t)batch[i] * F4 + c) * 4);
    atomicMax(pp + 0, __float_as_uint(v.x));
    atomicMax(pp + 1, __float_as_uint(v.y));
    atomicMax(pp + 2, __float_as_uint(v.z));
    atomicMax(pp + 3, __float_as_uint(v.w));
}

// ---------------- tail ----------------
__global__ void bias_add(float* __restrict__ y, const float* __restrict__ b,
                         int rows, int cols)
{
    int idx = blockIdx.x * blockDim.x + threadIdx.x;
    if (idx >= rows * cols) return;
    y[idx] += b[idx % cols];
}

// Row-wise L2 normalize [rows, 64] -> out
__global__ void l2_normalize(const float* __restrict__ y, float* __restrict__ out,
                             int cols)
{
    int r = blockIdx.x;
    int c = threadIdx.x;                  // blockDim.x == cols == 64
    __shared__ float sh[64];
    float v = y[(size_t)r * cols + c];
    sh[c] = v * v;
    __syncthreads();
    for (int st = cols >> 1; st > 0; st >>= 1) {
        if (c < st) sh[c] += sh[c + st];
        __syncthreads();
    }
    float nrm = fmaxf(sqrtf(sh[0]), 1e-12f);
    out[(size_t)r * cols + c] = v / nrm;
}

// ---------------------------------------------------------------------------
extern "C" void kernel_launch(void* const* d_in, const int* in_sizes, int n_in,
                              void* d_out, int out_size, void* d_ws, size_t ws_size,
                              hipStream_t stream)
{
    // Inputs (setup_inputs dict order):
    // 0:x 1:edge_index 2:batch, then per conv layer W,b,g,be (1..4),
    // W5,b5,g5,be5, W6,b6,g6,be6, W7,b7
    const float* x     = (const float*)d_in[0];
    const int*   ei    = (const int*)  d_in[1];
    const int*   batch = (const int*)  d_in[2];
    const int N = in_sizes[2];            // 20000 (multiple of 16)
    const int E = in_sizes[1] / 2;        // 256000
    const int G = 64;

    const int* esrc = ei;                 // edge_index[0]
    const int* edst = ei + E;             // edge_index[1]

    const float* W1 = (const float*)d_in[3];
    const float* g1 = (const float*)d_in[5];  const float* be1 = (const float*)d_in[6];
    const float* W2 = (const float*)d_in[7];
    const float* g2 = (const float*)d_in[9];  const float* be2 = (const float*)d_in[10];
    const float* W3 = (const float*)d_in[11];
    const float* g3 = (const float*)d_in[13]; const float* be3 = (const float*)d_in[14];
    const float* W4 = (const float*)d_in[15];
    const float* g4 = (const float*)d_in[17]; const float* be4 = (const float*)d_in[18];
    const float* W5 = (const float*)d_in[19];
    const float* g5 = (const float*)d_in[21]; const float* be5 = (const float*)d_in[22];
    const float* W6 = (const float*)d_in[23];
    const float* g6 = (const float*)d_in[25]; const float* be6 = (const float*)d_in[26];
    const float* W7 = (const float*)d_in[27];
    const float* b7 = (const float*)d_in[28];

    // Workspace layout (floats). Two N*1024 ping-pong buffers (~164 MB) + small.
    float* w = (float*)d_ws;
    float* bufA = w;  w += (size_t)N * 1024;  // GEMM output (lin)
    float* bufB = w;  w += (size_t)N * 1024;  // aggregation / layer features
    float* xpad = w;  w += (size_t)N * 8;     // x zero-padded to K=8
    float* w1p  = w;  w += 8 * 32;            // W1 zero-padded to [8,32]
    float* deg  = w;  w += N;
    float* dinv = w;  w += N;
    float* mean = w;  w += 1024;
    float* var  = w;  w += 1024;
    float* pool = w;  w += (size_t)G * 1024;
    float* t1   = w;  w += (size_t)G * 512;
    float* t2   = w;  w += (size_t)G * 256;
    float* t3   = w;  w += (size_t)G * 64;
    (void)ws_size; (void)n_in;

    const int TB = 256;
    auto cdiv = [](long long a, long long b) { return (unsigned)((a + b - 1) / b); };

    auto gemm = [&](const float* A, const float* Wp, float* C, int M, int K, int Nc) {
        dim3 blk(32, 4);
        dim3 grd(M / 16, ((Nc / 16) + 3) / 4);
        gemm_wmma_f32<<<grd, blk, 0, stream>>>(A, Wp, C, M, K, Nc);
    };

    // ---- pad x/W1; degrees & symmetric normalization ----
    pad_x     <<<cdiv((long long)N * 8, TB), TB, 0, stream>>>(x, xpad, N);
    pad_w1    <<<1, 256, 0, stream>>>(W1, w1p);
    deg_init  <<<cdiv(N, TB), TB, 0, stream>>>(deg, N);
    deg_count <<<cdiv(E, TB), TB, 0, stream>>>(edst, deg, E);
    deg_rsqrt <<<cdiv(N, TB), TB, 0, stream>>>(deg, dinv, N);

    // ---- GCN conv + BN + ReLU (output lands in bufB, in place) ----
    auto conv = [&](const float* hin, int Fin, const float* Wp,
                    const float* gp, const float* bep, int Fout) {
        const int F4 = Fout / 4;
        gemm(hin, Wp, bufA, N, Fin, Fout);
        selfloop_init4<<<cdiv((long long)N * F4, TB), TB, 0, stream>>>(
            (const float4*)bufA, deg, (float4*)bufB, N, F4);
        edge_scatter4 <<<cdiv((long long)E * F4, TB), TB, 0, stream>>>(
            esrc, edst, dinv, (const float4*)bufA, bufB, E, F4);
        bn_stats      <<<Fout, 256, 0, stream>>>(bufB, mean, var, N, Fout);
        bn_relu       <<<cdiv((long long)N * Fout, TB), TB, 0, stream>>>(
            bufB, mean, var, gp, bep, N, Fout);
    };

    conv(xpad, 8,    w1p, g1, be1, 32);   // padded: [N,8] @ [8,32]
    conv(bufB, 32,   W2,  g2, be2, 64);
    conv(bufB, 64,   W3,  g3, be3, 128);
    conv(bufB, 128,  W4,  g4, be4, 1024);

    // ---- global max pool [G, 1024] ----
    pool_zero<<<cdiv((long long)G * 1024, TB), TB, 0, stream>>>(pool, G * 1024);
    pool_max4<<<cdiv((long long)N * 256, TB), TB, 0, stream>>>(
        (const float4*)bufB, batch, pool, N, 256);

    // ---- FC head ----
    gemm(pool, W5, t1, G, 1024, 512);
    bn_stats<<<512, 256, 0, stream>>>(t1, mean, var, G, 512);
    bn_relu <<<cdiv((long long)G * 512, TB), TB, 0, stream>>>(t1, mean, var, g5, be5, G, 512);

    gemm(t1, W6, t2, G, 512, 256);
    bn_stats<<<256, 256, 0, stream>>>(t2, mean, var, G, 256);
    bn_relu <<<cdiv((long long)G * 256, TB), TB, 0, stream>>>(t2, mean, var, g6, be6, G, 256);

    gemm(t2, W7, t3, G, 256, 64);
    bias_add<<<cdiv((long long)G * 64, TB), TB, 0, stream>>>(t3, b7, G, 64);
    l2_normalize<<<G, 64, 0, stream>>>(t3, (float*)d_out, 64);
    (void)out_size;
}